// FAGCN_13374528159891
// MI455X (gfx1250) — compile-verified
//
#include <hip/hip_runtime.h>

typedef float v2f __attribute__((ext_vector_type(2)));
typedef float v8f __attribute__((ext_vector_type(8)));

#define EPS_F 0.3f

// ---------------- degree / norm ----------------

__global__ void fagcn_zero_u32(unsigned* __restrict__ p, int n) {
    int i = blockIdx.x * blockDim.x + threadIdx.x;
    if (i < n) p[i] = 0u;
}

__global__ void fagcn_degree(const int* __restrict__ dst, unsigned* __restrict__ deg, int E) {
    int i = blockIdx.x * blockDim.x + threadIdx.x;
    if (i < E) atomicAdd(&deg[dst[i]], 1u);
}

__global__ void fagcn_norm(const unsigned* __restrict__ deg, float* __restrict__ nrm, int n) {
    int i = blockIdx.x * blockDim.x + threadIdx.x;
    if (i < n) {
        float d = (float)deg[i];
        nrm[i] = rsqrtf(d < 1.0f ? 1.0f : d);
    }
}

// ---------------- h = relu(x @ w1^T + b1) via fp32 WMMA ----------------
// One wave per 16-row tile; 8 v8f accumulators = full 128 output columns.
// A(16x4 f32): lane half -> K pair, vgpr -> K within pair. B(4x16) mirrors.
// Both A and B per-lane loads are contiguous float2 at [k + 2*half].

__global__ void __launch_bounds__(256)
fagcn_gemm_wmma(const float* __restrict__ x, const float* __restrict__ w1,
                const float* __restrict__ b1, float* __restrict__ h,
                float* __restrict__ out, int N, int tiles) {
    int wave = (int)((blockIdx.x * blockDim.x + threadIdx.x) >> 5);
    int lane = threadIdx.x & 31;
    if (wave >= tiles) return;             // wave-uniform: EXEC stays all-1s below

    const int m0   = wave << 4;
    const int half = lane >> 4;            // 0 or 1
    const int lcol = lane & 15;

    int mrow = m0 + lcol;
    if (mrow >= N) mrow = N - 1;           // clamp address for ragged last tile
    const float* xrow = x + (size_t)mrow * 256 + 2 * half;

    v8f acc[8] = {};

    for (int kk = 0; kk < 256; kk += 4) {
        v2f a = *(const v2f*)(xrow + kk);
#pragma unroll
        for (int nt = 0; nt < 8; ++nt) {
            const float* wrow = w1 + (size_t)(nt * 16 + lcol) * 256 + 2 * half;
            v2f b = *(const v2f*)(wrow + kk);
            acc[nt] = __builtin_amdgcn_wmma_f32_16x16x4_f32(
                false, a, false, b, (short)0, acc[nt], false, false);
        }
    }

    // C/D layout: vgpr r -> rows (r, r+8) split across lane halves; col = lane&15.
#pragma unroll
    for (int nt = 0; nt < 8; ++nt) {
#pragma unroll
        for (int r = 0; r < 8; ++r) {
            int row = m0 + r + 8 * half;
            if (row < N) {
                int col = nt * 16 + lcol;
                float v = acc[nt][r] + b1[col];
                v = v > 0.0f ? v : 0.0f;   // relu
                size_t idx = (size_t)row * 128 + col;
                h[idx]   = v;
                out[idx] = EPS_F * v;      // seed out with EPS * raw
            }
        }
    }
}

// ---------------- per-node gate dots: a_dst = h.w_d, a_src = h.w_s ----------------

__global__ void __launch_bounds__(256)
fagcn_gate_dots(const float* __restrict__ h, const float* __restrict__ gate_w,
                float* __restrict__ a_dst, float* __restrict__ a_src, int N) {
    int node = (int)((blockIdx.x * blockDim.x + threadIdx.x) >> 5);
    int lane = threadIdx.x & 31;
    if (node >= N) return;

    const float4 hv = *(const float4*)(h + (size_t)node * 128 + lane * 4);
    const float4 wd = *(const float4*)(gate_w + lane * 4);
    const float4 wsv = *(const float4*)(gate_w + 128 + lane * 4);

    float pd = hv.x * wd.x + hv.y * wd.y + hv.z * wd.z + hv.w * wd.w;
    float ps = hv.x * wsv.x + hv.y * wsv.y + hv.z * wsv.z + hv.w * wsv.w;

#pragma unroll
    for (int off = 16; off > 0; off >>= 1) {   // wave32 reduction
        pd += __shfl_xor(pd, off);
        ps += __shfl_xor(ps, off);
    }
    if (lane == 0) { a_dst[node] = pd; a_src[node] = ps; }
}

// ---------------- edge scatter: out[dst] += h[src] * e ----------------
// One wave per edge; lane 0 evaluates the gate, broadcast via shfl;
// each lane handles 4 channels with hardware fp32 atomics (L2-resident).

__global__ void __launch_bounds__(256)
fagcn_edge(const int* __restrict__ ei, const float* __restrict__ h,
           const float* __restrict__ a_dst, const float* __restrict__ a_src,
           const float* __restrict__ nrm, const float* __restrict__ gate_b,
           float* __restrict__ out, int E) {
    int eid  = (int)((blockIdx.x * blockDim.x + threadIdx.x) >> 5);
    int lane = threadIdx.x & 31;
    if (eid >= E) return;

    int s = ei[eid];
    int d = ei[E + eid];

    float ev = 0.0f;
    if (lane == 0) {
        float g = tanhf(a_dst[d] + a_src[s] + gate_b[0]);
        ev = g * nrm[d] * nrm[s];
    }
    ev = __shfl(ev, 0);

    const float4 hv = *(const float4*)(h + (size_t)s * 128 + lane * 4);
    float* op = out + (size_t)d * 128 + lane * 4;
    unsafeAtomicAdd(op + 0, hv.x * ev);
    unsafeAtomicAdd(op + 1, hv.y * ev);
    unsafeAtomicAdd(op + 2, hv.z * ev);
    unsafeAtomicAdd(op + 3, hv.w * ev);
}

// ---------------- launcher ----------------

extern "C" void kernel_launch(void* const* d_in, const int* in_sizes, int n_in,
                              void* d_out, int out_size, void* d_ws, size_t ws_size,
                              hipStream_t stream) {
    const float* x      = (const float*)d_in[0];
    const int*   ei     = (const int*)d_in[1];
    const float* w1     = (const float*)d_in[2];
    const float* b1     = (const float*)d_in[3];
    const float* gate_w = (const float*)d_in[4];
    const float* gate_b = (const float*)d_in[5];

    const int N = in_sizes[0] / 256;
    const int E = in_sizes[1] / 2;

    float* out = (float*)d_out;

    // workspace layout
    float*    h     = (float*)d_ws;            // N*128
    float*    a_dst = h + (size_t)N * 128;     // N
    float*    a_src = a_dst + N;               // N
    float*    nrm   = a_src + N;               // N
    unsigned* deg   = (unsigned*)(nrm + N);    // N

    const int tiles = (N + 15) >> 4;

    fagcn_zero_u32<<<(N + 255) / 256, 256, 0, stream>>>(deg, N);
    fagcn_degree  <<<(E + 255) / 256, 256, 0, stream>>>(ei + E, deg, E);
    fagcn_norm    <<<(N + 255) / 256, 256, 0, stream>>>(deg, nrm, N);

    fagcn_gemm_wmma<<<(tiles + 7) / 8, 256, 0, stream>>>(x, w1, b1, h, out, N, tiles);

    fagcn_gate_dots<<<(N + 7) / 8, 256, 0, stream>>>(h, gate_w, a_dst, a_src, N);

    fagcn_edge<<<(E + 7) / 8, 256, 0, stream>>>(ei, h, a_dst, a_src, nrm, gate_b, out, E);
}